// PillarFeatureNet_214748364885
// MI455X (gfx1250) — compile-verified
//
#include <hip/hip_runtime.h>

// ---------------------------------------------------------------------------
// PillarFeatureNet for MI455X (gfx1250, wave32, WMMA).
// Pass A: per-pillar GEMM (feats 16x12 @ Wpad 12x64) via v_wmma_f32_16x16x4_f32,
//         producing per-pillar raw max/min of x (pre-BN) + deterministic
//         per-block partial sums for BN batch statistics.
// Pass B: reduce partials -> per-channel scale/shift (BN affine).
// Pass C: out = relu(scale * (scale>=0 ? max : min) + shift)   (monotonicity).
// ---------------------------------------------------------------------------

typedef float v8f __attribute__((ext_vector_type(8)));
typedef float v2f __attribute__((ext_vector_type(2)));

#define P_N    40000
#define M_N    100
#define M_PAD  112
#define C_OUT  64
#define WAVES  8
#define NBLK   (P_N / WAVES)          // 5000 blocks in pass A
#define BN_EPSf 1e-5f
#define VXf    0.16f
#define VYf    0.16f
#define X0f    0.0f
#define Y0f    (-39.68f)

// ---------------------------------------------------------------------------
// Pass A: one wave per pillar, 8 waves (256 threads) per block.
// WMMA layouts assumed per CDNA5 ISA 7.12.2:
//   A (16x4 f32):  lanes0-15 row M=lane {k0 in r0, k1 in r1}; lanes16-31 {k2,k3}
//   B (4x16 f32):  lanes hold col N=lane&15; lanes0-15 rows {k0,k1}, lanes16-31 {k2,k3}
//   C (16x16 f32): vgpr j: lanes0-15 -> row j, lanes16-31 -> row j+8; col = lane&15
// ---------------------------------------------------------------------------
__global__ __launch_bounds__(WAVES * 32) void pfn_gemm_wmma_kernel(
    const float* __restrict__ px, const float* __restrict__ py,
    const float* __restrict__ pz, const float* __restrict__ pi,
    const int* __restrict__ npts, const int* __restrict__ coors,
    const float* __restrict__ W,
    float* __restrict__ maxbuf, float* __restrict__ minbuf,
    float* __restrict__ partials)
{
  __shared__ float ldsX[WAVES][M_PAD];
  __shared__ float ldsY[WAVES][M_PAD];
  __shared__ float ldsZ[WAVES][M_PAD];
  __shared__ float ldsI[WAVES][M_PAD];
  __shared__ float wstat[WAVES][2 * C_OUT];   // per-wave {sum[64], sumsq[64]}

  const int tid  = threadIdx.x;
  const int wave = tid >> 5;
  const int lane = tid & 31;
  const int p    = blockIdx.x * WAVES + wave;   // pillar id (exact division)
  const int col  = lane & 15;
  const bool khi = lane >= 16;

  // ---- stage this pillar's points into LDS (zero-padded to 112) + sums ----
  float sx = 0.f, sy = 0.f, sz = 0.f;
  #pragma unroll
  for (int t = 0; t < 4; ++t) {
    int m = lane + 32 * t;
    if (m < M_PAD) {
      float vx = 0.f, vy = 0.f, vz = 0.f, vi = 0.f;
      if (m < M_N) {
        vx = px[p * M_N + m]; vy = py[p * M_N + m];
        vz = pz[p * M_N + m]; vi = pi[p * M_N + m];
      }
      ldsX[wave][m] = vx; ldsY[wave][m] = vy;
      ldsZ[wave][m] = vz; ldsI[wave][m] = vi;
      sx += vx; sy += vy; sz += vz;
    }
  }
  // wave32 butterfly reduction for per-pillar sums
  #pragma unroll
  for (int off = 16; off >= 1; off >>= 1) {
    sx += __shfl_xor(sx, off);
    sy += __shfl_xor(sy, off);
    sz += __shfl_xor(sz, off);
  }
  const int   n    = npts[p];
  const float invn = 1.0f / fmaxf((float)n, 1.0f);
  const float mx = sx * invn, my = sy * invn, mz = sz * invn;
  const float xb = (float)coors[4 * p + 3] * VXf + VXf * 0.5f + X0f;
  const float yb = (float)coors[4 * p + 2] * VYf + VYf * 0.5f + Y0f;

  // ---- B matrices: Wpad (12x64, rows 9..11 zero) -> 3 k-steps x 4 col-blocks
  v2f b[3][4];
  #pragma unroll
  for (int ks = 0; ks < 3; ++ks) {
    #pragma unroll
    for (int cb = 0; cb < 4; ++cb) {
      int r0 = ks * 4 + (khi ? 2 : 0);
      int c  = cb * 16 + col;
      v2f bb;
      bb.x = (r0     < 9) ? W[r0 * C_OUT + c]       : 0.0f;
      bb.y = (r0 + 1 < 9) ? W[(r0 + 1) * C_OUT + c] : 0.0f;
      b[ks][cb] = bb;
    }
  }

  __syncthreads();

  float maxv[4], minv[4], ssum[4], ssq[4];
  #pragma unroll
  for (int cb = 0; cb < 4; ++cb) {
    maxv[cb] = -__builtin_inff(); minv[cb] = __builtin_inff();
    ssum[cb] = 0.f; ssq[cb] = 0.f;
  }

  // ---- 7 point-tiles of 16 rows; 12 WMMAs per tile ----
  #pragma unroll 1
  for (int t = 0; t < 7; ++t) {
    int m = t * 16 + col;                  // same row for both lane halves
    float x  = ldsX[wave][m], y = ldsY[wave][m];
    float z  = ldsZ[wave][m], w4 = ldsI[wave][m];
    float mask = (m < n) ? 1.0f : 0.0f;    // n <= 99 => padded rows masked
    float f0 = x * mask, f1 = y * mask, f2 = z * mask, f3 = w4 * mask;
    float f4 = (x - mx) * mask, f5 = (y - my) * mask, f6 = (z - mz) * mask;
    float f7 = xb * mask, f8 = yb * mask;

    v2f a0, a1, a2;                        // per-lane-half k selection (v_cndmask)
    a0.x = khi ? f2 : f0;  a0.y = khi ? f3 : f1;
    a1.x = khi ? f6 : f4;  a1.y = khi ? f7 : f5;
    a2.x = khi ? 0.f : f8; a2.y = 0.f;     // k=9..11 are zero padding

    #pragma unroll
    for (int cb = 0; cb < 4; ++cb) {
      v8f acc = {0.f, 0.f, 0.f, 0.f, 0.f, 0.f, 0.f, 0.f};
      acc = __builtin_amdgcn_wmma_f32_16x16x4_f32(false, a0, false, b[0][cb],
                                                  (short)0, acc, false, false);
      acc = __builtin_amdgcn_wmma_f32_16x16x4_f32(false, a1, false, b[1][cb],
                                                  (short)0, acc, false, false);
      acc = __builtin_amdgcn_wmma_f32_16x16x4_f32(false, a2, false, b[2][cb],
                                                  (short)0, acc, false, false);
      float tmax = acc[0], tmin = acc[0], ts = 0.f, tq = 0.f;
      #pragma unroll
      for (int j = 0; j < 8; ++j) {
        float v = acc[j];
        tmax = fmaxf(tmax, v); tmin = fminf(tmin, v);
        ts += v; tq += v * v;
      }
      maxv[cb] = fmaxf(maxv[cb], tmax);
      minv[cb] = fminf(minv[cb], tmin);
      ssum[cb] += ts; ssq[cb] += tq;
    }
  }

  // combine lane halves (rows 0-7 vs 8-15 of each tile share a column)
  #pragma unroll
  for (int cb = 0; cb < 4; ++cb) {
    maxv[cb] = fmaxf(maxv[cb], __shfl_xor(maxv[cb], 16));
    minv[cb] = fminf(minv[cb], __shfl_xor(minv[cb], 16));
    ssum[cb] += __shfl_xor(ssum[cb], 16);
    ssq[cb]  += __shfl_xor(ssq[cb], 16);
  }

  if (lane < 16) {
    #pragma unroll
    for (int cb = 0; cb < 4; ++cb) {
      int d = cb * 16 + lane;
      maxbuf[p * C_OUT + d] = maxv[cb];
      minbuf[p * C_OUT + d] = minv[cb];
      wstat[wave][d]          = ssum[cb];
      wstat[wave][C_OUT + d]  = ssq[cb];
    }
  }
  __syncthreads();

  // deterministic per-block partials (no global atomics)
  if (tid < 2 * C_OUT) {
    float s = 0.f;
    #pragma unroll
    for (int w2 = 0; w2 < WAVES; ++w2) s += wstat[w2][tid];
    partials[blockIdx.x * (2 * C_OUT) + tid] = s;
  }
}

// ---------------------------------------------------------------------------
// Pass B: one block per channel; reduce NBLK partials, emit BN scale/shift.
// ---------------------------------------------------------------------------
__global__ __launch_bounds__(256) void pfn_stats_kernel(
    const float* __restrict__ partials,
    const float* __restrict__ gamma, const float* __restrict__ beta,
    float* __restrict__ scsh)
{
  const int d   = blockIdx.x;   // channel
  const int tid = threadIdx.x;
  __shared__ float ls[256], lq[256];
  float s = 0.f, q = 0.f;
  for (int i = tid; i < NBLK; i += 256) {
    s += partials[i * (2 * C_OUT) + d];
    q += partials[i * (2 * C_OUT) + C_OUT + d];
  }
  ls[tid] = s; lq[tid] = q;
  __syncthreads();
  #pragma unroll
  for (int off = 128; off >= 1; off >>= 1) {
    if (tid < off) { ls[tid] += ls[tid + off]; lq[tid] += lq[tid + off]; }
    __syncthreads();
  }
  if (tid == 0) {
    const float inv = 1.0f / (float)(P_N * M_N);
    float mu  = ls[0] * inv;
    float var = lq[0] * inv - mu * mu;
    float sc  = gamma[d] * rsqrtf(var + BN_EPSf);
    scsh[d]         = sc;
    scsh[C_OUT + d] = beta[d] - mu * sc;
  }
}

// ---------------------------------------------------------------------------
// Pass C: relu(affine) of max (or min for negative scale) -> out (P,64).
// ---------------------------------------------------------------------------
__global__ __launch_bounds__(256) void pfn_out_kernel(
    const float* __restrict__ maxbuf, const float* __restrict__ minbuf,
    const float* __restrict__ scsh, float* __restrict__ out)
{
  int idx = blockIdx.x * blockDim.x + threadIdx.x;
  if (idx >= P_N * C_OUT) return;
  int d = idx & (C_OUT - 1);
  float sc = scsh[d], sh = scsh[C_OUT + d];
  float v  = (sc >= 0.f) ? maxbuf[idx] : minbuf[idx];
  out[idx] = fmaxf(__builtin_fmaf(sc, v, sh), 0.0f);
}

// ---------------------------------------------------------------------------
extern "C" void kernel_launch(void* const* d_in, const int* in_sizes, int n_in,
                              void* d_out, int out_size, void* d_ws, size_t ws_size,
                              hipStream_t stream)
{
  (void)in_sizes; (void)n_in; (void)out_size; (void)ws_size;
  const float* px    = (const float*)d_in[0];
  const float* py    = (const float*)d_in[1];
  const float* pz    = (const float*)d_in[2];
  const float* pi    = (const float*)d_in[3];
  const int*   npts  = (const int*)d_in[4];
  const int*   coors = (const int*)d_in[5];
  const float* W     = (const float*)d_in[6];
  const float* gamma = (const float*)d_in[7];
  const float* beta  = (const float*)d_in[8];
  float* out = (float*)d_out;

  float* ws       = (float*)d_ws;
  float* maxbuf   = ws;                                   // P*64
  float* minbuf   = ws + (size_t)P_N * C_OUT;             // P*64
  float* partials = ws + (size_t)2 * P_N * C_OUT;         // NBLK*128
  float* scsh     = partials + (size_t)NBLK * 2 * C_OUT;  // 128

  pfn_gemm_wmma_kernel<<<NBLK, WAVES * 32, 0, stream>>>(
      px, py, pz, pi, npts, coors, W, maxbuf, minbuf, partials);
  pfn_stats_kernel<<<C_OUT, 256, 0, stream>>>(partials, gamma, beta, scsh);
  pfn_out_kernel<<<(P_N * C_OUT + 255) / 256, 256, 0, stream>>>(
      maxbuf, minbuf, scsh, out);
}